// MultiHeadAttention_738734374918
// MI455X (gfx1250) — compile-verified
//
#include <hip/hip_runtime.h>
#include <hip/hip_bf16.h>
#include <math.h>

// Problem constants
#define BB    2
#define LL    2048
#define DD    1024
#define HH    16
#define DHD   64
#define BHN   (BB * HH)      // 32
#define MROWS (BB * LL)      // 4096
#define MT    (MROWS / 16)   // 256 row tiles
#define JT    (LL / 16)      // 128 key/query tiles per (b,h)

typedef float v2f __attribute__((ext_vector_type(2)));
typedef float v8f __attribute__((ext_vector_type(8)));

// D = A(16x4, f32) * B(4x16, f32) + C(16x16, f32)
// A: lane l holds A[l%16][2*(l/16) + {0,1}]
// B: lane l holds B[2*(l/16) + {0,1}][l%16]
// C/D: VGPR v, lane l -> row v + 8*(l/16), col l%16
__device__ __forceinline__ v8f wmma_f32(v2f a, v2f b, v8f c) {
  return __builtin_amdgcn_wmma_f32_16x16x4_f32(false, a, false, b, (short)0, c,
                                               false, false);
}

// ---------------------------------------------------------------------------
// Kernel 1: Q/K/V projections. One wave -> 16 rows x 64 cols (full DH).
// grid waves = MT * HH * 3. Scale 1/sqrt(DH) folded into Q.
// ---------------------------------------------------------------------------
__global__ __launch_bounds__(128) void qkv_proj_kernel(
    const float* __restrict__ x,
    const float* __restrict__ Wq, const float* __restrict__ bq,
    const float* __restrict__ Wk, const float* __restrict__ bk,
    const float* __restrict__ Wv, const float* __restrict__ bv,
    float* __restrict__ Qo, float* __restrict__ Ko, float* __restrict__ Vo) {
  const int lane = threadIdx.x & 31;
  const int gw   = (blockIdx.x * blockDim.x + threadIdx.x) >> 5;
  const int mt   = gw / (HH * 3);
  const int rem  = gw % (HH * 3);
  const int h    = rem / 3;
  const int t    = rem % 3;
  const int r    = lane & 15;
  const int half = lane >> 4;

  const float* W    = (t == 0 ? Wq : (t == 1 ? Wk : Wv)) + (size_t)h * DD * DHD;
  const float* bias = (t == 0 ? bq : (t == 1 ? bk : bv)) + h * DHD;
  float*       Out  = (t == 0 ? Qo : (t == 1 ? Ko : Vo));

  const int    m0   = mt * 16;
  const float* arow = x + (size_t)(m0 + r) * DD;

  v8f acc[4] = {};
  for (int k0 = 0; k0 < DD; k0 += 4) {
    const int koff = k0 + 2 * half;
    v2f a = *(const v2f*)(arow + koff);                 // contiguous b64 load
    const float* w0 = W + (size_t)koff * DHD;
    const float* w1 = w0 + DHD;
#pragma unroll
    for (int nb = 0; nb < 4; ++nb) {
      v2f b;
      b.x = w0[nb * 16 + r];
      b.y = w1[nb * 16 + r];
      acc[nb] = wmma_f32(a, b, acc[nb]);
    }
  }

  const float sc = (t == 0) ? 0.125f : 1.0f;  // 1/sqrt(64) folded into Q
#pragma unroll
  for (int nb = 0; nb < 4; ++nb) {
#pragma unroll
    for (int v = 0; v < 8; ++v) {
      const int row = m0 + v + 8 * half;
      const int bb  = row / LL;
      const int l   = row % LL;
      const int n   = nb * 16 + r;
      Out[(((size_t)bb * HH + h) * LL + l) * DHD + n] =
          (acc[nb][v] + bias[n]) * sc;
    }
  }
}

// ---------------------------------------------------------------------------
// Kernel 2: column softmax stats. Softmax is over the QUERY axis (i), so for
// each column j we need m[j] = max_i s[i,j], l[j] = sum_i exp(s[i,j]-m[j]).
// One wave owns a 16-column tile, streams all 2048 rows through WMMA.
// ---------------------------------------------------------------------------
__global__ __launch_bounds__(128) void col_stats_kernel(
    const float* __restrict__ Q, const float* __restrict__ K,
    float* __restrict__ Mb, float* __restrict__ Lb) {
  const int lane = threadIdx.x & 31;
  const int gw   = (blockIdx.x * blockDim.x + threadIdx.x) >> 5;
  const int bh   = gw / JT;
  const int jt   = gw % JT;
  const int r    = lane & 15;
  const int half = lane >> 4;

  const float* Qb = Q + (size_t)bh * LL * DHD;
  const float* Kb = K + (size_t)bh * LL * DHD;

  // Hoisted K^T fragments for this 16-column tile (16 k-steps of 4)
  v2f bf[16];
#pragma unroll
  for (int ks = 0; ks < 16; ++ks)
    bf[ks] = *(const v2f*)(Kb + (size_t)(jt * 16 + r) * DHD + 4 * ks + 2 * half);

  float m = -INFINITY, l = 0.0f;
  for (int it = 0; it < JT; ++it) {
    v8f s = {};
    const float* qrow = Qb + (size_t)(it * 16 + r) * DHD + 2 * half;
#pragma unroll
    for (int ks = 0; ks < 16; ++ks) {
      v2f a = *(const v2f*)(qrow + 4 * ks);
      s = wmma_f32(a, bf[ks], s);
    }
#pragma unroll
    for (int v = 0; v < 8; ++v) {   // online max/sumexp over 8 rows per lane
      const float xv = s[v];
      const float mn = fmaxf(m, xv);
      l = l * __expf(m - mn) + __expf(xv - mn);
      m = mn;
    }
  }
  // merge the two half-wave row groups (lane j and lane j+16 hold column j)
  const float m2 = __shfl_xor(m, 16, 32);
  const float l2 = __shfl_xor(l, 16, 32);
  const float M  = fmaxf(m, m2);
  const float Ls = l * __expf(m - M) + l2 * __expf(m2 - M);
  if (lane < 16) {
    Mb[(size_t)bh * LL + jt * 16 + lane] = M;
    Lb[(size_t)bh * LL + jt * 16 + lane] = Ls;
  }
}

// ---------------------------------------------------------------------------
// Kernel 3: out = P * V where P[i,j] = exp(s[i,j]-m[j]) / l[j].
// One wave -> 16 query rows x 64 (DH). Recompute S tiles via WMMA, normalize,
// bounce P through padded LDS to reshape C-layout -> A-fragment layout,
// then accumulate P*V with a second set of WMMAs.
// ---------------------------------------------------------------------------
__global__ __launch_bounds__(128) void attn_out_kernel(
    const float* __restrict__ Q, const float* __restrict__ K,
    const float* __restrict__ V, const float* __restrict__ Mb,
    const float* __restrict__ Lb, float* __restrict__ A) {
  __shared__ float Pt[4][16][20];  // per-wave 16x16, padded -> conflict-free
  const int lane = threadIdx.x & 31;
  const int wib  = threadIdx.x >> 5;
  const int gw   = (blockIdx.x * blockDim.x + threadIdx.x) >> 5;
  const int bh   = gw / JT;
  const int it   = gw % JT;
  const int r    = lane & 15;
  const int half = lane >> 4;
  const int b    = bh / HH;
  const int h    = bh % HH;

  const float* Qb   = Q + (size_t)bh * LL * DHD;
  const float* Kb   = K + (size_t)bh * LL * DHD;
  const float* Vb   = V + (size_t)bh * LL * DHD;
  const float* mrow = Mb + (size_t)bh * LL;
  const float* lrow = Lb + (size_t)bh * LL;

  // Hoisted Q fragments for this 16-row tile
  v2f qa[16];
#pragma unroll
  for (int ks = 0; ks < 16; ++ks)
    qa[ks] = *(const v2f*)(Qb + (size_t)(it * 16 + r) * DHD + 4 * ks + 2 * half);

  v8f acc[4] = {};
  float(*__restrict__ P)[20] = Pt[wib];

  for (int jt = 0; jt < JT; ++jt) {
    v8f s = {};
    const float* krow = Kb + (size_t)(jt * 16 + r) * DHD + 2 * half;
#pragma unroll
    for (int ks = 0; ks < 16; ++ks) {
      v2f bfr = *(const v2f*)(krow + 4 * ks);
      s = wmma_f32(qa[ks], bfr, s);
    }
    const int   j  = jt * 16 + r;          // this lane's column
    const float mj = mrow[j];
    const float rl = 1.0f / lrow[j];
#pragma unroll
    for (int v = 0; v < 8; ++v)            // C-layout -> LDS row-major
      P[v + 8 * half][r] = __expf(s[v] - mj) * rl;
    // wave-private LDS; DS ops are in-order per wave, compiler inserts waits
#pragma unroll
    for (int ks2 = 0; ks2 < 4; ++ks2) {
      v2f a;
      a.x = P[r][4 * ks2 + 2 * half];
      a.y = P[r][4 * ks2 + 2 * half + 1];
      const float* v0 = Vb + (size_t)(jt * 16 + 4 * ks2 + 2 * half) * DHD;
      const float* v1 = v0 + DHD;
#pragma unroll
      for (int nb = 0; nb < 4; ++nb) {
        v2f bv;
        bv.x = v0[nb * 16 + r];
        bv.y = v1[nb * 16 + r];
        acc[nb] = wmma_f32(a, bv, acc[nb]);
      }
    }
  }
  // store head-concatenated: A[b][l][h*64 + n]
#pragma unroll
  for (int nb = 0; nb < 4; ++nb)
#pragma unroll
    for (int v = 0; v < 8; ++v) {
      const int l = it * 16 + v + 8 * half;
      A[((size_t)b * LL + l) * DD + h * DHD + nb * 16 + r] = acc[nb][v];
    }
}

// ---------------------------------------------------------------------------
// Kernel 4: Y = A @ WO^T + bO  (WO is [out,in], so B fragments are contiguous)
// ---------------------------------------------------------------------------
__global__ __launch_bounds__(128) void out_proj_kernel(
    const float* __restrict__ A, const float* __restrict__ WO,
    const float* __restrict__ bO, float* __restrict__ Y) {
  const int lane = threadIdx.x & 31;
  const int gw   = (blockIdx.x * blockDim.x + threadIdx.x) >> 5;
  const int NT   = DD / 64;  // 16
  const int mt   = gw / NT;
  const int nt   = gw % NT;
  const int r    = lane & 15;
  const int half = lane >> 4;

  const float* arow = A + (size_t)(mt * 16 + r) * DD;

  v8f acc[4] = {};
  for (int k0 = 0; k0 < DD; k0 += 4) {
    const int koff = k0 + 2 * half;
    v2f a = *(const v2f*)(arow + koff);
#pragma unroll
    for (int nb = 0; nb < 4; ++nb) {
      const int e = nt * 64 + nb * 16 + r;
      v2f bfr = *(const v2f*)(WO + (size_t)e * DD + koff);  // WO[e][k],WO[e][k+1]
      acc[nb] = wmma_f32(a, bfr, acc[nb]);
    }
  }
#pragma unroll
  for (int nb = 0; nb < 4; ++nb)
#pragma unroll
    for (int v = 0; v < 8; ++v) {
      const int row = mt * 16 + v + 8 * half;
      const int e   = nt * 64 + nb * 16 + r;
      Y[(size_t)row * DD + e] = acc[nb][v] + bO[e];
    }
}

// ---------------------------------------------------------------------------
extern "C" void kernel_launch(void* const* d_in, const int* in_sizes, int n_in,
                              void* d_out, int out_size, void* d_ws,
                              size_t ws_size, hipStream_t stream) {
  const float* x  = (const float*)d_in[0];
  const float* Wq = (const float*)d_in[1];
  const float* bq = (const float*)d_in[2];
  const float* Wk = (const float*)d_in[3];
  const float* bk = (const float*)d_in[4];
  const float* Wv = (const float*)d_in[5];
  const float* bv = (const float*)d_in[6];
  const float* WO = (const float*)d_in[7];
  const float* bO = (const float*)d_in[8];
  float* out = (float*)d_out;
  float* ws  = (float*)d_ws;

  const size_t QSZ = (size_t)BB * HH * LL * DHD;  // 4M floats
  float* Qd = ws;
  float* Kd = ws + QSZ;
  float* Vd = ws + 2 * QSZ;
  float* Ad = ws + 3 * QSZ;                       // attn output [B,L,D]
  float* Md = ws + 4 * QSZ;                       // column max [B*H,L]
  float* Ld = Md + (size_t)BB * HH * LL;          // column sumexp [B*H,L]

  {
    const int waves = MT * HH * 3;  // 12288 waves, 4 per block
    qkv_proj_kernel<<<dim3(waves / 4), dim3(128), 0, stream>>>(
        x, Wq, bq, Wk, bk, Wv, bv, Qd, Kd, Vd);
  }
  {
    const int waves = BHN * JT;  // 4096
    col_stats_kernel<<<dim3(waves / 4), dim3(128), 0, stream>>>(Qd, Kd, Md, Ld);
  }
  {
    const int waves = BHN * JT;  // 4096
    attn_out_kernel<<<dim3(waves / 4), dim3(128), 0, stream>>>(Qd, Kd, Vd, Md,
                                                               Ld, Ad);
  }
  {
    const int waves = MT * (DD / 64);  // 4096
    out_proj_kernel<<<dim3(waves / 4), dim3(128), 0, stream>>>(Ad, WO, bO, out);
  }
}